// GATConv_86672440033518
// MI455X (gfx1250) — compile-verified
//
#include <hip/hip_runtime.h>
#include <math.h>

typedef __attribute__((ext_vector_type(2))) float v2f;
typedef __attribute__((ext_vector_type(8))) float v8f;

#define NN 50000
#define NE 800000
#define IN_DIM 256
#define HH 4
#define DD 32
#define HD 128        // H*D
#define NEG_SLOPE 0.2f

// ---------- order-preserving float<->uint mapping for atomic max ----------
__device__ __forceinline__ unsigned f2ord(float f) {
    unsigned u = __float_as_uint(f);
    return (u & 0x80000000u) ? ~u : (u | 0x80000000u);
}
__device__ __forceinline__ float ord2f(unsigned u) {
    return __uint_as_float((u & 0x80000000u) ? (u & 0x7FFFFFFFu) : ~u);
}

// ---------- K0: init rst accumulator, emax (-inf encoded), denom ----------
__global__ void init_kernel(float* __restrict__ rst, unsigned* __restrict__ emaxU,
                            float* __restrict__ denom) {
    int i = blockIdx.x * blockDim.x + threadIdx.x;
    if (i < NN * HD) rst[i] = 0.0f;
    if (i < NN * HH) { emaxU[i] = 0x007FFFFFu; denom[i] = 0.0f; } // ~bits(-inf)
}

// ---------- K1: ft = feat @ W via V_WMMA_F32_16X16X4_F32 ----------
// One wave computes a 16-row x 128-col strip: 8 C tiles, K=256 in steps of 4.
__global__ __launch_bounds__(256)
void gemm_ft_kernel(const float* __restrict__ feat, const float* __restrict__ W,
                    float* __restrict__ ft) {
    const int lane    = threadIdx.x & 31;
    const int wid     = threadIdx.x >> 5;
    const int gw      = blockIdx.x * 8 + wid;       // 16-row block id (wave-uniform)
    if (gw >= NN / 16) return;                      // whole-wave exit: EXEC stays all-1
    const int rowBase = gw * 16;
    const int halfSel = lane >> 4;                  // 0: K pair {0,1}, 1: {2,3}
    const int laneM   = lane & 15;

    v8f acc[8];
    #pragma unroll
    for (int t = 0; t < 8; ++t)
        acc[t] = (v8f){0.f,0.f,0.f,0.f,0.f,0.f,0.f,0.f};

    // A fragment pointer: row = rowBase+laneM, cols k0 + 2*halfSel .. +1
    const float* aRow = feat + (size_t)(rowBase + laneM) * IN_DIM + 2 * halfSel;

    for (int k0 = 0; k0 < IN_DIM; k0 += 4) {
        const float2 av = *(const float2*)(aRow + k0);
        v2f a; a.x = av.x; a.y = av.y;
        const int kr = k0 + 2 * halfSel;
        const float* wRow0 = W + (size_t)kr * HD + laneM;
        #pragma unroll
        for (int t = 0; t < 8; ++t) {
            v2f b;
            b.x = wRow0[t * 16];            // B[kr    ][t*16+laneM]
            b.y = wRow0[HD + t * 16];       // B[kr + 1][t*16+laneM]
            acc[t] = __builtin_amdgcn_wmma_f32_16x16x4_f32(
                false, a, false, b, (short)0, acc[t], false, false);
        }
    }

    // C/D layout: VGPR r -> row rowBase + r + 8*halfSel, col t*16 + laneM
    #pragma unroll
    for (int t = 0; t < 8; ++t) {
        const int col = t * 16 + laneM;
        #pragma unroll
        for (int r = 0; r < 8; ++r) {
            const int row = rowBase + r + 8 * halfSel;
            ft[(size_t)row * HD + col] = acc[t][r];
        }
    }
}

// ---------- K2: per-(node,head) scalars, z reparam, outputs 2 & 3 ----------
__global__ void node_scalar_kernel(const float* __restrict__ ft,
                                   const float* __restrict__ mu_src,
                                   const float* __restrict__ mu_dst,
                                   const float* __restrict__ lam_src,
                                   const float* __restrict__ eps_src,
                                   const float* __restrict__ eps_dst,
                                   float* __restrict__ z_src, float* __restrict__ z_dst,
                                   float* __restrict__ out_mu, float* __restrict__ out_lam) {
    int i = blockIdx.x * blockDim.x + threadIdx.x;   // i = n*H + h
    if (i >= NN * HH) return;
    const int h = i & (HH - 1);
    const float* f   = ft + (size_t)i * DD;          // ft[n,h,:] is contiguous
    const float* mus = mu_src + h * DD;
    const float* mud = mu_dst + h * DD;
    const float* las = lam_src + h * DD;
    float ms = 0.f, md = 0.f, ls = 0.f;
    #pragma unroll
    for (int d = 0; d < DD; d += 4) {
        float4 fv = *(const float4*)(f + d);
        float4 m1 = *(const float4*)(mus + d);
        float4 m2 = *(const float4*)(mud + d);
        float4 l1 = *(const float4*)(las + d);
        ms += fv.x*m1.x + fv.y*m1.y + fv.z*m1.z + fv.w*m1.w;
        md += fv.x*m2.x + fv.y*m2.y + fv.z*m2.z + fv.w*m2.w;
        ls += fv.x*l1.x + fv.y*l1.y + fv.z*l1.z + fv.w*l1.w;
    }
    const float sdev = expf(0.5f * ls);              // lam_d == lam_s (ref uses lam_src twice)
    z_src[i]  = eps_src[i] * sdev + ms;
    z_dst[i]  = eps_dst[i] * sdev + md;
    out_mu[i] = ms + md;
    out_lam[i] = 2.0f * ls;
}

__device__ __forceinline__ float edge_logit(const int* src, const int* dst,
                                            const float* ppmi, const float* z_src,
                                            const float* z_dst, int e, int h,
                                            int& d_out_idx) {
    const int s = src[e], d = dst[e];
    d_out_idx = d * HH + h;
    float v = z_src[s * HH + h] + z_dst[d_out_idx];
    v = (v >= 0.f) ? v : NEG_SLOPE * v;
    float lp = fmaxf(logf(ppmi[e]), 1.0f);
    float sc = fmaxf(logf(lp), 1.0f);
    return v * sc;
}

// ---------- K3: segment max over dst ----------
__global__ void edge_max_kernel(const int* __restrict__ src, const int* __restrict__ dst,
                                const float* __restrict__ ppmi,
                                const float* __restrict__ z_src, const float* __restrict__ z_dst,
                                unsigned* __restrict__ emaxU) {
    int i = blockIdx.x * blockDim.x + threadIdx.x;   // i = e*H + h
    if (i >= NE * HH) return;
    int di;
    float v = edge_logit(src, dst, ppmi, z_src, z_dst, i >> 2, i & 3, di);
    atomicMax(&emaxU[di], f2ord(v));
}

// ---------- K4: segment sum of exp ----------
__global__ void edge_sum_kernel(const int* __restrict__ src, const int* __restrict__ dst,
                                const float* __restrict__ ppmi,
                                const float* __restrict__ z_src, const float* __restrict__ z_dst,
                                const unsigned* __restrict__ emaxU, float* __restrict__ denom) {
    int i = blockIdx.x * blockDim.x + threadIdx.x;
    if (i >= NE * HH) return;
    int di;
    float v = edge_logit(src, dst, ppmi, z_src, z_dst, i >> 2, i & 3, di);
    atomicAdd(&denom[di], expf(v - ord2f(emaxU[di])));
}

// ---------- K5: weighted scatter-sum; one wave per (edge,head), lane = d ----------
__global__ __launch_bounds__(256)
void edge_aggr_kernel(const int* __restrict__ src, const int* __restrict__ dst,
                      const float* __restrict__ ppmi,
                      const float* __restrict__ z_src, const float* __restrict__ z_dst,
                      const unsigned* __restrict__ emaxU, const float* __restrict__ denom,
                      const float* __restrict__ ft, float* __restrict__ rst) {
    const long long tid = (long long)blockIdx.x * blockDim.x + threadIdx.x;
    const int w    = (int)(tid >> 5);                // wave index = e*H + h
    const int lane = (int)(tid & 31);                // d
    if (w >= NE * HH) return;
    const int e = w >> 2, h = w & 3;
    int di;
    const float v = edge_logit(src, dst, ppmi, z_src, z_dst, e, h, di);
    const float a = expf(v - ord2f(emaxU[di])) / denom[di];
    const int s = src[e];
    const float fval = ft[(size_t)(s * HH + h) * DD + lane];
    atomicAdd(&rst[(size_t)di * DD + lane], fval * a);
}

extern "C" void kernel_launch(void* const* d_in, const int* in_sizes, int n_in,
                              void* d_out, int out_size, void* d_ws, size_t ws_size,
                              hipStream_t stream) {
    const float* feat    = (const float*)d_in[0];
    const int*   src     = (const int*)  d_in[1];
    const int*   dst     = (const int*)  d_in[2];
    const float* ppmi    = (const float*)d_in[3];
    const float* eps_src = (const float*)d_in[4];
    const float* eps_dst = (const float*)d_in[5];
    const float* W       = (const float*)d_in[6];
    const float* mu_src  = (const float*)d_in[7];
    const float* mu_dst  = (const float*)d_in[8];
    const float* lam_src = (const float*)d_in[9];
    // d_in[10] = lam_dst (unused by reference math)

    // outputs: rst [N*H*D] | mu_s+mu_d [N*H] | lam_s+lam_d [N*H]
    float* rst     = (float*)d_out;
    float* out_mu  = rst + (size_t)NN * HD;
    float* out_lam = out_mu + (size_t)NN * HH;

    // workspace: ft | z_src | z_dst | denom | emaxU
    float*    ws    = (float*)d_ws;
    float*    ft    = ws;                                   // N*128
    float*    z_src = ft + (size_t)NN * HD;                 // N*4
    float*    z_dst = z_src + (size_t)NN * HH;              // N*4
    float*    denom = z_dst + (size_t)NN * HH;              // N*4
    unsigned* emaxU = (unsigned*)(denom + (size_t)NN * HH); // N*4

    const int TPB = 256;

    init_kernel<<<(NN * HD + TPB - 1) / TPB, TPB, 0, stream>>>(rst, emaxU, denom);

    const int nRowBlocks = NN / 16;                          // 3125
    gemm_ft_kernel<<<(nRowBlocks + 7) / 8, TPB, 0, stream>>>(feat, W, ft);

    node_scalar_kernel<<<(NN * HH + TPB - 1) / TPB, TPB, 0, stream>>>(
        ft, mu_src, mu_dst, lam_src, eps_src, eps_dst, z_src, z_dst, out_mu, out_lam);

    const int nEH = NE * HH;
    edge_max_kernel<<<(nEH + TPB - 1) / TPB, TPB, 0, stream>>>(
        src, dst, ppmi, z_src, z_dst, emaxU);
    edge_sum_kernel<<<(nEH + TPB - 1) / TPB, TPB, 0, stream>>>(
        src, dst, ppmi, z_src, z_dst, emaxU, denom);

    const long long aggrThreads = (long long)nEH * 32;
    edge_aggr_kernel<<<(unsigned)((aggrThreads + TPB - 1) / TPB), TPB, 0, stream>>>(
        src, dst, ppmi, z_src, z_dst, emaxU, denom, ft, rst);
}